// GCN_25975962206307
// MI455X (gfx1250) — compile-verified
//
#include <hip/hip_runtime.h>
#include <hip/hip_bf16.h>
#include <stdint.h>
#include <math.h>

typedef float v2f __attribute__((ext_vector_type(2)));
typedef float v8f __attribute__((ext_vector_type(8)));

// ---------------------------------------------------------------------------
// Device-scope non-returning float atomic max -> GLOBAL_ATOMIC_MAX_NUM_F32.
// Tracked by STOREcnt; S_ENDPGM performs an implicit wait-idle, so no explicit
// s_wait is needed before kernel exit.
// ---------------------------------------------------------------------------
__device__ __forceinline__ void atomic_max_f32(float* addr, float v) {
    asm volatile("global_atomic_max_num_f32 %0, %1, off scope:SCOPE_DEV"
                 :: "v"(addr), "v"(v)
                 : "memory");
}

// ---------------------------------------------------------------------------
// Linear layer 1: h = x @ w1^T + b1.   x:[N,128], w1:[32,128], h:[N,32]
// One wave (32 threads) per 16-row block; V_WMMA_F32_16X16X4_F32, K swept by 4.
// A frag (16x4 f32, 2 VGPRs): lanes 0-15 -> K={k,k+1}, lanes 16-31 -> K={k+2,k+3}.
// B frag (4x16 f32, 2 VGPRs): mirrored layout, N = lane%16.
// ---------------------------------------------------------------------------
__global__ void lin1_wmma(const float* __restrict__ x,
                          const float* __restrict__ w1,
                          const float* __restrict__ b1,
                          float* __restrict__ h) {
    const int rb   = blockIdx.x;          // 16-row block
    const int lane = threadIdx.x;         // 0..31, full wave => EXEC all-1s
    const int half = lane >> 4;           // 0: K-lo pair, 1: K-hi pair
    const int l16  = lane & 15;

    const float* xrow  = x  + (size_t)(rb * 16 + l16) * 128 + 2 * half; // A row
    const float* w0row = w1 + (size_t)(l16)      * 128 + 2 * half;      // B, n=0..15
    const float* w1row = w1 + (size_t)(16 + l16) * 128 + 2 * half;      // B, n=16..31

    v8f c0 = {};
    v8f c1 = {};
    for (int k = 0; k < 128; k += 4) {
        v2f a  = *(const v2f*)(xrow  + k);
        v2f bA = *(const v2f*)(w0row + k);
        v2f bB = *(const v2f*)(w1row + k);
        c0 = __builtin_amdgcn_wmma_f32_16x16x4_f32(false, a, false, bA,
                                                   (short)0, c0, false, false);
        c1 = __builtin_amdgcn_wmma_f32_16x16x4_f32(false, a, false, bB,
                                                   (short)0, c1, false, false);
    }

    // D layout: VGPR v -> M = v (lanes 0-15) / v+8 (lanes 16-31); N = lane%16.
    const float bias0 = b1[l16];
    const float bias1 = b1[16 + l16];
    #pragma unroll
    for (int v = 0; v < 8; ++v) {
        const size_t m = (size_t)rb * 16 + v + 8 * half;
        h[m * 32 + l16]      = c0[v] + bias0;
        h[m * 32 + 16 + l16] = c1[v] + bias1;
    }
}

// ---------------------------------------------------------------------------
// Linear layer 2 with fused ReLU/isolated-node fixup on the INPUT:
//   h2 = max(agg1, 0) @ w2^T + b2.     agg1:[N,32] (may hold -inf), w2:[32,32]
// fmaxf(-inf, 0) == 0 reproduces jnp.where(isneginf, 0, agg) followed by relu.
// ---------------------------------------------------------------------------
__global__ void lin2_wmma(const float* __restrict__ agg1,
                          const float* __restrict__ w2,
                          const float* __restrict__ b2,
                          float* __restrict__ h2) {
    const int rb   = blockIdx.x;
    const int lane = threadIdx.x;
    const int half = lane >> 4;
    const int l16  = lane & 15;

    const float* arow  = agg1 + (size_t)(rb * 16 + l16) * 32 + 2 * half;
    const float* w0row = w2 + (size_t)(l16)      * 32 + 2 * half;
    const float* w1row = w2 + (size_t)(16 + l16) * 32 + 2 * half;

    v8f c0 = {};
    v8f c1 = {};
    #pragma unroll
    for (int k = 0; k < 32; k += 4) {
        v2f a;
        a.x = fmaxf(arow[k], 0.0f);       // fused ReLU + (-inf -> 0)
        a.y = fmaxf(arow[k + 1], 0.0f);
        v2f bA = *(const v2f*)(w0row + k);
        v2f bB = *(const v2f*)(w1row + k);
        c0 = __builtin_amdgcn_wmma_f32_16x16x4_f32(false, a, false, bA,
                                                   (short)0, c0, false, false);
        c1 = __builtin_amdgcn_wmma_f32_16x16x4_f32(false, a, false, bB,
                                                   (short)0, c1, false, false);
    }

    const float bias0 = b2[l16];
    const float bias1 = b2[16 + l16];
    #pragma unroll
    for (int v = 0; v < 8; ++v) {
        const size_t m = (size_t)rb * 16 + v + 8 * half;
        h2[m * 32 + l16]      = c0[v] + bias0;
        h2[m * 32 + 16 + l16] = c1[v] + bias1;
    }
}

// ---------------------------------------------------------------------------
// Edge scatter-max: one lane per (edge, feature). A wave covers one edge's 32
// features -> coalesced 128B gather of h[src], 32 f32 atomic-max to agg[dst].
// ---------------------------------------------------------------------------
__global__ void scatter_max(const float* __restrict__ h,
                            const long long* __restrict__ src,
                            const long long* __restrict__ dst,
                            float* __restrict__ agg,
                            long long E) {
    const long long idx = (long long)blockIdx.x * blockDim.x + threadIdx.x;
    const long long e = idx >> 5;
    const int f = (int)(idx & 31);
    if (e >= E) return;
    const long long s = src[e];
    const long long d = dst[e];
    const float v = h[s * 32 + f];
    atomic_max_f32(agg + d * 32 + f, v);
}

__global__ void fill_neg_inf(float* __restrict__ p, int n) {
    const int i = blockIdx.x * blockDim.x + threadIdx.x;
    if (i < n) p[i] = -INFINITY;
}

__global__ void fixup_neg_inf(float* __restrict__ p, int n) {
    const int i = blockIdx.x * blockDim.x + threadIdx.x;
    if (i < n) {
        const float v = p[i];
        p[i] = (v == -INFINITY) ? 0.0f : v;
    }
}

// ---------------------------------------------------------------------------
// Orchestration. Inputs: x[f32 N*128], edge_index[i64 2*E], w1[32,128],
// b1[32], w2[32,32], b2[32]. Output: f32 N*32.
// Workspace: buf (h1, later reused for h2) + agg1  => 2 * N*32 f32.
// ---------------------------------------------------------------------------
extern "C" void kernel_launch(void* const* d_in, const int* in_sizes, int n_in,
                              void* d_out, int out_size, void* d_ws, size_t ws_size,
                              hipStream_t stream) {
    const float*     x  = (const float*)d_in[0];
    const long long* ei = (const long long*)d_in[1];
    const float*     w1 = (const float*)d_in[2];
    const float*     b1 = (const float*)d_in[3];
    const float*     w2 = (const float*)d_in[4];
    const float*     b2 = (const float*)d_in[5];

    const int       N   = in_sizes[0] / 128;       // 100000 (divisible by 16)
    const long long E   = in_sizes[1] / 2;         // 1600000
    const long long* src = ei;                     // edge_index[0]
    const long long* dst = ei + E;                 // edge_index[1]

    float* buf  = (float*)d_ws;                    // h1, then h2
    float* agg1 = buf + (size_t)N * 32;            // conv1 aggregation
    float* out  = (float*)d_out;                   // conv2 aggregation / result

    const int rowBlocks = N / 16;                  // 6250 waves of WMMA work
    const int nFeat     = N * 32;                  // 3.2M
    const int fillGrid  = (nFeat + 255) / 256;
    const long long scatterThreads = E * 32;       // 51.2M lanes
    const int scatterGrid = (int)((scatterThreads + 255) / 256);

    // conv1
    lin1_wmma<<<rowBlocks, 32, 0, stream>>>(x, w1, b1, buf);
    fill_neg_inf<<<fillGrid, 256, 0, stream>>>(agg1, nFeat);
    scatter_max<<<scatterGrid, 256, 0, stream>>>(buf, src, dst, agg1, E);

    // conv2 (ReLU + isolated-node fixup fused into operand load)
    lin2_wmma<<<rowBlocks, 32, 0, stream>>>(agg1, w2, b2, buf);
    fill_neg_inf<<<fillGrid, 256, 0, stream>>>(out, nFeat);
    scatter_max<<<scatterGrid, 256, 0, stream>>>(buf, src, dst, out, E);
    fixup_neg_inf<<<fillGrid, 256, 0, stream>>>(out, nFeat);
}